// FlattenedWindowMapping_46505905881637
// MI455X (gfx1250) — compile-verified
//
#include <hip/hip_runtime.h>
#include <hip/hip_bf16.h>

// ---------------------------------------------------------------------------
// FlattenedWindowMapping for MI455X (gfx1250, wave32).
// Memory-bound integer workload: two stable LSD radix argsorts (32-bit keys)
// + per-batch padding maps. The 256-bin radix-histogram prefix scans are done
// on the matrix pipe via V_WMMA_F32_16X16X4_F32:
//   P_incl = Lstrict x (H x Ones) + H x Uincl   (exact in f32, counts < 2^24)
// ---------------------------------------------------------------------------

typedef float v2f __attribute__((ext_vector_type(2)));
typedef float v8f __attribute__((ext_vector_type(8)));

#define GSZ   128
#define BATCH 16
// meta layout (ints)
#define M_NUM   0   // 16
#define M_BS    16  // 17
#define M_NUMP  33  // 16
#define M_BSP   49  // 17  (meta[M_BSP+16] == n_p)
#define M_BIAS  66  // 16
#define META_INTS 128

#define ITEMS_PER_BLOCK 4096   // 256 threads x 16 items

// ---------------------------------------------------------------- batch meta
__global__ void k_zero_meta(int* meta) {
  if (threadIdx.x < META_INTS) meta[threadIdx.x] = 0;
}

__global__ void k_count(const int* __restrict__ coords, int n, int* meta) {
  int i = blockIdx.x * blockDim.x + threadIdx.x;
  if (i < n) atomicAdd(&meta[M_NUM + coords[4 * i]], 1);
}

__global__ void k_batchscan(int* meta) {
  if (threadIdx.x == 0 && blockIdx.x == 0) {
    meta[M_BS] = 0; meta[M_BSP] = 0;
    for (int b = 0; b < BATCH; ++b) {
      int nb = meta[M_NUM + b];
      int np = ((nb + GSZ - 1) / GSZ) * GSZ;
      meta[M_NUMP + b]    = np;
      meta[M_BS  + b + 1] = meta[M_BS  + b] + nb;
      meta[M_BSP + b + 1] = meta[M_BSP + b] + np;
      meta[M_BIAS + b]    = meta[M_BSP + b] - meta[M_BS + b];
    }
  }
}

// ------------------------------------------------------------- map kernels
__global__ void k_win2flat(const int* __restrict__ coords, int n,
                           const int* __restrict__ meta, float* __restrict__ out) {
  int i = blockIdx.x * blockDim.x + threadIdx.x;
  if (i >= n) return;
  int np = meta[M_BSP + BATCH];
  int b  = coords[4 * i];
  out[np + i] = (float)(i + meta[M_BIAS + b]);
}

__global__ void k_flat2win(const int* __restrict__ meta, float* __restrict__ out) {
  int k = blockIdx.x * blockDim.x + threadIdx.x;
  int np = meta[M_BSP + BATCH];
  if (k >= np) return;
  int b = 0;
  for (int j = 0; j < BATCH - 1; ++j)
    if (k >= meta[M_BSP + j + 1]) b = j + 1;    // searchsorted(right)-1
  int bs  = meta[M_BS + b],  bsp = meta[M_BSP + b];
  int nb  = meta[M_NUM + b], npb = meta[M_NUMP + b];
  int bias = bsp - bs;
  bool in_tail = (nb != npb) && (k >= bsp + nb);
  bool multi   = (npb != GSZ);
  int denom = nb > 1 ? nb : 1;
  int tail  = multi ? (k - GSZ - bias) : (bs + ((k - bsp - nb) % denom));
  out[k] = (float)(in_tail ? tail : (k - bias));
}

// mx=ceil(1024/16)+1=65, my=65, mz=ceil(32/8)+1=5; m_per_sample=21125
// win_vol = 16*16*8 = 2048
__global__ void k_keys(const int* __restrict__ coords, int n, int axis,
                       int* __restrict__ key, int* __restrict__ idx) {
  int i = blockIdx.x * blockDim.x + threadIdx.x;
  if (i >= n) return;
  int b = coords[4*i], z = coords[4*i+1], y = coords[4*i+2], x = coords[4*i+3];
  int wcx = x >> 4, wcy = y >> 4, wcz = z >> 3;
  int cix = x & 15, ciy = y & 15, ciz = z & 7;
  int k;
  if (axis == 0) {
    int bwin = b * 21125 + wcx * 325 + wcy * 5 + wcz;   // my*mz = 325
    k = bwin * 2048 + cix * 128 + ciy * 8 + ciz;        // wy*wz=128, wz=8
  } else {
    int bwin = b * 21125 + wcy * 325 + wcx * 5 + wcz;   // mx*mz = 325
    k = bwin * 2048 + ciy * 128 + cix * 8 + ciz;        // wx*wz=128
  }
  key[i] = k; idx[i] = i;
}

// ------------------------------------------------------------- radix: hist
__global__ void k_hist(const int* __restrict__ keys, int n, int shift,
                       int* __restrict__ hist, int NB) {
  __shared__ int lh[256];
  int tid = threadIdx.x;
  lh[tid] = 0;
  __syncthreads();
  int base = blockIdx.x * ITEMS_PER_BLOCK;
  #pragma unroll 4
  for (int c = 0; c < 16; ++c) {
    int i = base + c * 256 + tid;
    __builtin_prefetch(keys + i + 512, 0, 3);           // global_prefetch_b8
    if (i < n) atomicAdd(&lh[((unsigned)keys[i] >> shift) & 255u], 1);
  }
  __syncthreads();
  hist[tid * NB + blockIdx.x] = lh[tid];                // digit-major layout
}

// ------------------------------------------- WMMA 256-element exclusive scan
__device__ __forceinline__ v8f wmma4(v2f a, v2f b, v8f c) {
  return __builtin_amdgcn_wmma_f32_16x16x4_f32(false, a, false, b,
                                               (short)0, c, false, false);
}

// lds: 256 floats (row-major 16x16, bin = 16*i + j). Output e[8] in C-layout
// (lane l, vgpr v -> element (v + 8*(l>>4), l&15)), total = sum of all 256.
__device__ __forceinline__ void wmma_scan256(const float* lds, int lane,
                                             float* e, float* totalOut) {
  int m = lane & 15, hi = lane >> 4;
  v8f acc = {0,0,0,0,0,0,0,0};
  v8f g   = {0,0,0,0,0,0,0,0};
  #pragma unroll
  for (int t = 0; t < 4; ++t) {                 // K = 16 as 4 chunks of 4
    int k0 = 4*t + (hi ? 2 : 0);
    int k1 = 4*t + (hi ? 3 : 1);
    v2f a; a.x = lds[m*16 + k0]; a.y = lds[m*16 + k1];   // H, A-layout
    v2f u; u.x = (k0 <= m) ? 1.f : 0.f;                  // Uincl, B-layout
           u.y = (k1 <= m) ? 1.f : 0.f;
    acc = wmma4(a, u, acc);                              // += H x Uincl
    v2f ones; ones.x = 1.f; ones.y = 1.f;
    g = wmma4(a, ones, g);                               // G = H x Ones
  }
  float R[16];                                           // row sums R[i]
  #pragma unroll
  for (int k = 0; k < 16; ++k) {
    float gv = (k < 8) ? g[k] : g[k - 8];
    R[k] = __shfl(gv, (k < 8) ? 0 : 16, 32);
  }
  #pragma unroll
  for (int t = 0; t < 4; ++t) {                          // += Lstrict x G
    int k0 = 4*t + (hi ? 2 : 0);
    int k1 = 4*t + (hi ? 3 : 1);
    v2f a; a.x = (k0 < m) ? 1.f : 0.f; a.y = (k1 < m) ? 1.f : 0.f;
    v2f b; b.x = R[k0]; b.y = R[k1];
    acc = wmma4(a, b, acc);
  }
  #pragma unroll
  for (int v = 0; v < 8; ++v)                            // inclusive -> exclusive
    e[v] = acc[v] - lds[(v + 8*hi) * 16 + m];
  *totalOut = __shfl(acc[7], 31, 32);                    // P[15][15]
}

__global__ void k_scan_chunks(int* __restrict__ hist, int* __restrict__ chunkSums,
                              int nElems) {
  __shared__ float lds[256];
  int lane = threadIdx.x;
  int base = blockIdx.x * 256;
  for (int i = lane; i < 256; i += 32) {
    int gi = base + i;
    lds[i] = (gi < nElems) ? (float)hist[gi] : 0.f;
  }
  __syncthreads();
  float e[8], total;
  wmma_scan256(lds, lane, e, &total);
  int m = lane & 15, hi = lane >> 4;
  #pragma unroll
  for (int v = 0; v < 8; ++v) {
    int gi = base + (v + 8*hi) * 16 + m;
    if (gi < nElems) hist[gi] = (int)e[v];
  }
  if (lane == 0) chunkSums[blockIdx.x] = (int)total;
}

__global__ void k_scan_top(int* __restrict__ sums, int nElems) {
  __shared__ float lds[256];
  int lane = threadIdx.x;
  int m = lane & 15, hi = lane >> 4;
  float carry = 0.f;
  int nChunks = (nElems + 255) / 256;
  for (int c = 0; c < nChunks; ++c) {
    int base = c * 256;
    for (int i = lane; i < 256; i += 32) {
      int gi = base + i;
      lds[i] = (gi < nElems) ? (float)sums[gi] : 0.f;
    }
    __syncthreads();
    float e[8], total;
    wmma_scan256(lds, lane, e, &total);
    #pragma unroll
    for (int v = 0; v < 8; ++v) {
      int gi = base + (v + 8*hi) * 16 + m;
      if (gi < nElems) sums[gi] = (int)(e[v] + carry);
    }
    carry += total;
    __syncthreads();
  }
}

__global__ void k_add(int* __restrict__ hist, const int* __restrict__ sums, int nElems) {
  int i = blockIdx.x * blockDim.x + threadIdx.x;
  if (i < nElems) hist[i] += sums[i >> 8];
}

// ----------------------------------------------------- stable radix scatter
__global__ void k_scatter(const int* __restrict__ keysIn, const int* __restrict__ idxIn,
                          int n, int shift, const int* __restrict__ scanHist, int NB,
                          int* __restrict__ keysOut, int* __restrict__ idxOut) {
  __shared__ int waveHist[8][257];
  __shared__ int blockCount[257];
  int tid = threadIdx.x;
  int w = tid >> 5, lane = tid & 31;
  for (int i = tid; i < 257; i += 256) blockCount[i] = 0;
  for (int i = tid; i < 8 * 257; i += 256) (&waveHist[0][0])[i] = 0;
  __syncthreads();

  int base = blockIdx.x * ITEMS_PER_BLOCK;
  for (int c = 0; c < 16; ++c) {
    int i = base + c * 256 + tid;
    bool valid = (i < n);
    int key = 0, payload = 0;
    unsigned d9 = 256u;                          // 9-bit sentinel for invalid
    if (valid) {
      key = keysIn[i]; payload = idxIn[i];
      d9 = ((unsigned)key >> shift) & 255u;
    }
    // wave32 digit match (8 data bits + 1 validity bit)
    unsigned peers = 0xffffffffu;
    #pragma unroll
    for (int bit = 0; bit < 9; ++bit) {
      unsigned b = (d9 >> bit) & 1u;
      unsigned vote = (unsigned)__ballot((int)b);
      peers &= b ? vote : ~vote;
    }
    unsigned rank = __popc(peers & ((1u << lane) - 1u));
    int cnt = __popc(peers);
    if (rank == 0) waveHist[w][d9] = cnt;
    __syncthreads();
    if (valid) {
      int off = blockCount[d9];
      for (int w2 = 0; w2 < w; ++w2) off += waveHist[w2][d9];
      int pos = scanHist[(int)d9 * NB + blockIdx.x] + off + (int)rank;
      keysOut[pos] = key;
      idxOut[pos]  = payload;
    }
    __syncthreads();
    for (int i2 = tid; i2 < 257; i2 += 256) {
      int s = 0;
      #pragma unroll
      for (int w2 = 0; w2 < 8; ++w2) { s += waveHist[w2][i2]; waveHist[w2][i2] = 0; }
      blockCount[i2] += s;
    }
    __syncthreads();
  }
}

__global__ void k_writeidx(const int* __restrict__ idx, int n,
                           const int* __restrict__ meta, int axis,
                           float* __restrict__ out) {
  int i = blockIdx.x * blockDim.x + threadIdx.x;
  if (i >= n) return;
  int np = meta[M_BSP + BATCH];
  out[np + n + axis * n + i] = (float)idx[i];
}

// ------------------------------------------------------------------- driver
extern "C" void kernel_launch(void* const* d_in, const int* in_sizes, int n_in,
                              void* d_out, int out_size, void* d_ws, size_t ws_size,
                              hipStream_t stream) {
  const int* coords = (const int*)d_in[0];
  int n = in_sizes[0] / 4;
  float* out = (float*)d_out;
  int* ws = (int*)d_ws;

  int* meta = ws;
  int* keyA = ws + META_INTS;
  int* idxA = keyA + n;
  int* keyB = idxA + n;
  int* idxB = keyB + n;
  int NB = (n + ITEMS_PER_BLOCK - 1) / ITEMS_PER_BLOCK;
  int histElems = NB * 256;
  int* hist = idxB + n;
  int* chunkSums = hist + histElems;

  dim3 b256(256), b32(32);
  int gN = (n + 255) / 256;

  k_zero_meta<<<1, 128, 0, stream>>>(meta);
  k_count<<<gN, b256, 0, stream>>>(coords, n, meta);
  k_batchscan<<<1, 32, 0, stream>>>(meta);
  k_win2flat<<<gN, b256, 0, stream>>>(coords, n, meta, out);
  int npmax = n + BATCH * GSZ;
  k_flat2win<<<(npmax + 255) / 256, b256, 0, stream>>>(meta, out);

  for (int axis = 0; axis < 2; ++axis) {
    k_keys<<<gN, b256, 0, stream>>>(coords, n, axis, keyA, idxA);
    int *kin = keyA, *iin = idxA, *kout = keyB, *iout = idxB;
    for (int p = 0; p < 4; ++p) {
      k_hist<<<NB, b256, 0, stream>>>(kin, n, 8 * p, hist, NB);
      k_scan_chunks<<<NB, b32, 0, stream>>>(hist, chunkSums, histElems);
      k_scan_top<<<1, b32, 0, stream>>>(chunkSums, NB);
      k_add<<<(histElems + 255) / 256, b256, 0, stream>>>(hist, chunkSums, histElems);
      k_scatter<<<NB, b256, 0, stream>>>(kin, iin, n, 8 * p, hist, NB, kout, iout);
      int* t;
      t = kin; kin = kout; kout = t;
      t = iin; iin = iout; iout = t;
    }
    // after 4 passes (even number of swaps) results are back in keyA/idxA == kin/iin
    k_writeidx<<<gN, b256, 0, stream>>>(iin, n, meta, axis, out);
  }
}